// BatchMultiHeadGraphAttention_67491116089371
// MI455X (gfx1250) — compile-verified
//
#include <hip/hip_runtime.h>
#include <hip/hip_bf16.h>
#include <float.h>
#include <math.h>

typedef __attribute__((ext_vector_type(2))) float v2f;
typedef __attribute__((ext_vector_type(8))) float v8f;

#define F_IN   256
#define F_OUT  64
#define NHEAD  4
#define HF     (NHEAD * F_OUT)   // 256 total output columns
#define NEG_SLOPE 0.2f

// ---------- monotone float<->uint encoding so atomicMax(u32) orders floats ---
__device__ __forceinline__ unsigned fenc(float f) {
  unsigned u = __float_as_uint(f);
  return (u & 0x80000000u) ? ~u : (u | 0x80000000u);
}
__device__ __forceinline__ float fdec(unsigned u) {
  return __uint_as_float((u & 0x80000000u) ? (u ^ 0x80000000u) : ~u);
}

// =====================  1) GEMM: HP = h @ Wflat  (N x 256)  ==================
// Wflat[k][h*64+o] = w[h][k][o].  One wave -> one 16x16 output tile.
// V_WMMA_F32_16X16X4_F32, K marched 4 at a time (64 WMMAs per tile).
__global__ void __launch_bounds__(128) gat_gemm_wmma(
    const float* __restrict__ A,    // N x F_IN, row major
    const float* __restrict__ W,    // [H][F_IN][F_OUT]
    float* __restrict__ HP,         // N x HF, row major
    int N)
{
  const int wid   = threadIdx.x >> 5;
  const int lane  = threadIdx.x & 31;
  const int m0    = blockIdx.x * 16;
  const int n0    = (blockIdx.y * 4 + wid) * 16;
  const int row   = lane & 15;     // A row within tile / B column within tile
  const int khalf = lane >> 4;     // 0: K=0,1   1: K=2,3

  const int head  = n0 >> 6;                       // all 16 cols of a tile share a head
  const float* Arow = A + (long long)(m0 + row) * F_IN + 2 * khalf;
  const float* Bcol = W + head * (F_IN * F_OUT) + (2 * khalf) * F_OUT + (n0 & 63) + row;

  // speculative prefetch of the next M-tile's A rows (dropped if OOB)
  __builtin_prefetch(Arow + 16 * F_IN, 0, 1);

  v8f c = {};
#pragma unroll 8
  for (int k = 0; k < F_IN; k += 4) {
    v2f a = *(const v2f*)(Arow + k);               // A[m][k+2*khalf .. +1]
    v2f b;
    b.x = Bcol[k * F_OUT];                         // B[k+2*khalf  ][n]
    b.y = Bcol[k * F_OUT + F_OUT];                 // B[k+2*khalf+1][n]
    c = __builtin_amdgcn_wmma_f32_16x16x4_f32(false, a, false, b,
                                              (short)0, c, false, false);
  }

  // C/D layout: VGPR v, lanes 0-15 -> M=v, lanes 16-31 -> M=8+v; N = lane%16
  const int ccol = n0 + (lane & 15);
  const int crow = m0 + (lane >> 4) * 8;
  float* out = HP + (long long)crow * HF + ccol;
#pragma unroll
  for (int v = 0; v < 8; ++v) out[(long long)v * HF] = c[v];
}

// =====================  2) attention logits per node  ========================
// s_src[h][n] = HP[n, h*64: h*64+64] . a_src[h],  s_dst likewise.
__global__ void gat_scores(const float* __restrict__ HP,
                           const float* __restrict__ fc,   // [H][128][1]
                           float* __restrict__ S,          // [8][N]: 0..3 src, 4..7 dst
                           int N)
{
  int idx = blockIdx.x * blockDim.x + threadIdx.x;
  if (idx >= N * NHEAD) return;
  int n = idx >> 2, h = idx & 3;
  const float* hp   = HP + (long long)n * HF + h * F_OUT;
  const float* asrc = fc + h * (2 * F_OUT);
  const float* adst = asrc + F_OUT;
  float ss = 0.f, sd = 0.f;
#pragma unroll 8
  for (int o = 0; o < F_OUT; ++o) { float v = hp[o]; ss += v * asrc[o]; sd += v * adst[o]; }
  S[(long long)h * N + n]           = ss;
  S[(long long)(4 + h) * N + n]     = sd;
}

// =====================  init scratch  =======================================
__global__ void gat_init(unsigned* __restrict__ segmax, float* __restrict__ den,
                         float* __restrict__ acc, int N)
{
  long long i = (long long)blockIdx.x * blockDim.x + threadIdx.x;
  if (i < (long long)NHEAD * N) { segmax[i] = 0u; den[i] = 0.f; }
  if (i < (long long)N * HF)    acc[i] = 0.f;
}

// =====================  3a) segment max over src  ============================
__global__ void gat_edge_max(const long long* __restrict__ ei,
                             const float* __restrict__ S,
                             unsigned* __restrict__ segmax,
                             int N, long long E)
{
  long long j = (long long)blockIdx.x * blockDim.x + threadIdx.x;
  if (j >= E * NHEAD) return;
  int h = (int)(j & 3);
  long long e = j >> 2;
  int src = (int)ei[e], dst = (int)ei[E + e];
  float v = S[(long long)h * N + src] + S[(long long)(4 + h) * N + dst];
  v = v > 0.f ? v : NEG_SLOPE * v;
  atomicMax(&segmax[(long long)h * N + src], fenc(v));
}

// =====================  3b) exp + segment sum  ===============================
__global__ void gat_edge_exp(const long long* __restrict__ ei,
                             const float* __restrict__ S,
                             const unsigned* __restrict__ segmax,
                             float* __restrict__ ex,     // [E][4]
                             float* __restrict__ den,    // [H][N]
                             int N, long long E)
{
  long long j = (long long)blockIdx.x * blockDim.x + threadIdx.x;
  if (j >= E * NHEAD) return;
  int h = (int)(j & 3);
  long long e = j >> 2;
  int src = (int)ei[e], dst = (int)ei[E + e];
  float v = S[(long long)h * N + src] + S[(long long)(4 + h) * N + dst];
  v = v > 0.f ? v : NEG_SLOPE * v;
  float x = expf(v - fdec(segmax[(long long)h * N + src]));
  ex[e * NHEAD + h] = x;
  atomicAdd(&den[(long long)h * N + src], x);
}

// =====================  3c) normalize + scatter messages  ====================
// One 256-thread block per edge: thread t covers head t/64, column t.
__global__ void __launch_bounds__(256) gat_scatter(
    const long long* __restrict__ ei,
    const float* __restrict__ ex,
    const float* __restrict__ den,
    const float* __restrict__ HP,
    float* __restrict__ acc,       // N x HF
    int N, long long E)
{
  long long e = blockIdx.x;
  int t = threadIdx.x;
  int h = t >> 6;
  int src = (int)ei[e], dst = (int)ei[E + e];
  float alpha = ex[e * NHEAD + h] / den[(long long)h * N + src];
  float m = alpha * HP[(long long)dst * HF + t];
  atomicAdd(&acc[(long long)src * HF + t], m);
}

// =====================  4) mean over heads + bias  ===========================
__global__ void gat_finalize(const float* __restrict__ acc,
                             const float* __restrict__ bias,
                             float* __restrict__ out, int N)
{
  int idx = blockIdx.x * blockDim.x + threadIdx.x;
  if (idx >= N * F_OUT) return;
  int n = idx >> 6, o = idx & 63;
  const float* a = acc + (long long)n * HF + o;
  float s = a[0] + a[F_OUT] + a[2 * F_OUT] + a[3 * F_OUT];
  out[idx] = 0.25f * s + bias[o];
}

extern "C" void kernel_launch(void* const* d_in, const int* in_sizes, int n_in,
                              void* d_out, int out_size, void* d_ws, size_t ws_size,
                              hipStream_t stream)
{
  const float*     h    = (const float*)d_in[0];
  const long long* ei   = (const long long*)d_in[1];   // int64 per reference
  const float*     w    = (const float*)d_in[2];
  const float*     fc   = (const float*)d_in[3];
  const float*     bias = (const float*)d_in[4];
  float*           out  = (float*)d_out;

  const int       N = in_sizes[0] / F_IN;
  const long long E = (long long)in_sizes[1] / 2;

  // workspace carve-up
  float*    HP     = (float*)d_ws;                                    // N*HF
  float*    S      = HP + (size_t)N * HF;                             // 8*N
  unsigned* SEGMAX = (unsigned*)(S + (size_t)8 * N);                  // 4*N
  float*    DEN    = (float*)(SEGMAX + (size_t)NHEAD * N);            // 4*N
  float*    EX     = DEN + (size_t)NHEAD * N;                         // E*4
  float*    ACC    = EX + (size_t)E * NHEAD;                          // N*HF

  // 1) WMMA GEMM (N divisible by 16: 50000 = 3125*16)
  dim3 gg(N / 16, (HF / 16) / 4);
  gat_gemm_wmma<<<gg, 128, 0, stream>>>(h, w, HP, N);

  // 2) per-node attention logits
  int t2 = (N * NHEAD + 255) / 256;
  gat_scores<<<t2, 256, 0, stream>>>(HP, fc, S, N);

  // init segmax/denom/acc
  long long initn = (long long)N * HF;
  gat_init<<<(unsigned)((initn + 255) / 256), 256, 0, stream>>>(SEGMAX, DEN, ACC, N);

  // 3) segment softmax over src + scatter
  long long ew = E * NHEAD;
  unsigned eb = (unsigned)((ew + 255) / 256);
  gat_edge_max<<<eb, 256, 0, stream>>>(ei, S, SEGMAX, N, E);
  gat_edge_exp<<<eb, 256, 0, stream>>>(ei, S, SEGMAX, EX, DEN, N, E);
  gat_scatter<<<(unsigned)E, 256, 0, stream>>>(ei, EX, DEN, HP, ACC, N, E);

  // 4) head mean + bias
  int t4 = (N * F_OUT + 255) / 256;
  gat_finalize<<<t4, 256, 0, stream>>>(ACC, bias, out, N);
}